// CoGOLLoss_59055800320712
// MI455X (gfx1250) — compile-verified
//
#include <hip/hip_runtime.h>
#include <hip/hip_bf16.h>

typedef float v2f __attribute__((ext_vector_type(2)));
typedef float v8f __attribute__((ext_vector_type(8)));
typedef float vf4 __attribute__((ext_vector_type(4)));

#define ALPHA_F 0.01f
#define BETA_F  0.05f
#define KM1 63
#define MAIN_BLOCKS 2048
#define REG_BLOCKS  256
#define THREADS     256

// ---------------------------------------------------------------------------
// Wave-level f32 sum using the matrix unit:
// D = ones(16x4) x B(4x16) + 0. B's two VGPRs hold {partial, 0} per lane, so
// the 64 B slots are exactly the 32 lane partials + 32 zeros. Each of the 16
// column sums of D is therefore a disjoint pair of lane partials covering all
// 32 lanes; d[0] at lane n holds colsum(n%16). Four xor-shuffles over the 16
// columns finish the reduction (all lanes end with the wave total).
// Stays entirely in f32 -> no precision loss vs. a shuffle tree.
// ---------------------------------------------------------------------------
__device__ __forceinline__ float wave_reduce32(float p) {
    v2f a; a.x = 1.0f; a.y = 1.0f;     // A = ones(16x4)
    v2f b; b.x = p;    b.y = 0.0f;     // B = lane partials + zero pad
    v8f c = {};
    v8f d = __builtin_amdgcn_wmma_f32_16x16x4_f32(
        /*neg_a=*/false, a, /*neg_b=*/false, b,
        /*c_mod=*/(short)0, c, /*reuse_a=*/false, /*reuse_b=*/false);
    float s = d[0];
    s += __shfl_xor(s, 1, 32);
    s += __shfl_xor(s, 2, 32);
    s += __shfl_xor(s, 4, 32);
    s += __shfl_xor(s, 8, 32);
    return s;
}

// Deterministic block reduction (8 waves @ 256 threads). Result on thread 0.
__device__ __forceinline__ float block_reduce(float p, float* lds) {
    float w = wave_reduce32(p);
    const int lane = threadIdx.x & 31;
    const int wid  = threadIdx.x >> 5;
    if (lane == 0) lds[wid] = w;
    __syncthreads();
    float s = 0.0f;
    if (threadIdx.x == 0) {
        const int nw = blockDim.x >> 5;
        for (int i = 0; i < nw; ++i) s += lds[i];   // fixed order
    }
    __syncthreads();  // make lds reusable by a second call
    return s;
}

// Numerically stable log_sigmoid(y) = min(y,0) - log1p(exp(-|y|)),
// using hardware v_exp_f32 / v_log_f32 (arg in (1,2], no cancellation).
__device__ __forceinline__ float log_sigmoid_f(float y) {
    float z = __expf(-fabsf(y));
    return fminf(y, 0.0f) - __logf(1.0f + z);
}

// ---------------------------------------------------------------------------
// Kernel 1: main loss reduction over logits (B x 63), streamed as float4.
// B*63 is divisible by 4, so no tail. Non-temporal loads keep the 132 MB
// stream out of L2 so the reused 2 MB targets array stays resident.
// ---------------------------------------------------------------------------
__global__ void __launch_bounds__(THREADS)
loss_kernel(const float* __restrict__ logits, const int* __restrict__ targets,
            float* __restrict__ partial, int B) {
    __shared__ float lds[THREADS / 32];
    const vf4* l4 = (const vf4*)logits;
    const unsigned N4     = (unsigned)B * (unsigned)KM1 / 4u;
    const unsigned stride = gridDim.x * blockDim.x;
    const float NEG_LN2 = -0.6931471805599453f;

    float acc = 0.0f;
    for (unsigned i = blockIdx.x * blockDim.x + threadIdx.x; i < N4; i += stride) {
        __builtin_prefetch(&l4[i + stride], 0, 0);     // global_prefetch_b8
        vf4 v = __builtin_nontemporal_load(&l4[i]);
        float x[4] = {v.x, v.y, v.z, v.w};
        const unsigned e = i * 4u;
#pragma unroll
        for (int u = 0; u < 4; ++u) {
            const unsigned idx = e + (unsigned)u;
            const unsigned row = idx / KM1;            // mul-hi by constant
            const int      j   = (int)(idx - row * KM1);   // j = c-1
            const int      t   = targets[row];
            if (j == 0 && t >= 2) acc += NEG_LN2;      // c = 0 term, once/row
            if (j <= t - 3) {                          // c <= t-2 : negative branch
                acc += log_sigmoid_f(-x[u]);
            } else if (j >= t - 1 && j <= KM1 - 2) {   // t <= c <= 62 : positive
                acc += log_sigmoid_f(x[u]);
            }                                          // j == t-2, j == 62: skip
        }
    }
    float s = block_reduce(acc, lds);
    if (threadIdx.x == 0) partial[blockIdx.x] = s;
}

// ---------------------------------------------------------------------------
// Kernel 2: ridge terms, pre-scaled: 0.5*ALPHA*sum(W^2) + 0.5*BETA*sum(deltas[1:]^2)
// ---------------------------------------------------------------------------
__global__ void __launch_bounds__(THREADS)
reg_kernel(const float* __restrict__ weights, int nw4,
           const float* __restrict__ deltas, float* __restrict__ partial) {
    __shared__ float lds[THREADS / 32];
    const vf4* w4 = (const vf4*)weights;
    const unsigned stride = gridDim.x * blockDim.x;

    float acc = 0.0f;
    for (unsigned i = blockIdx.x * blockDim.x + threadIdx.x; i < (unsigned)nw4; i += stride) {
        vf4 v = __builtin_nontemporal_load(&w4[i]);
        acc += (0.5f * ALPHA_F) * (v.x * v.x + v.y * v.y + v.z * v.z + v.w * v.w);
    }
    if (blockIdx.x == 0) {
        // deltas is (4, 63); rows 1..3 -> flat elements [63, 252)
        for (unsigned i = threadIdx.x; i < 3u * KM1; i += blockDim.x) {
            float d = deltas[KM1 + i];
            acc += (0.5f * BETA_F) * d * d;
        }
    }
    float s = block_reduce(acc, lds);
    if (threadIdx.x == 0) partial[blockIdx.x] = s;
}

// ---------------------------------------------------------------------------
// Kernel 3: fixed-order reduction of partials; out = -L/B + R
// ---------------------------------------------------------------------------
__global__ void __launch_bounds__(THREADS)
final_kernel(const float* __restrict__ lossPart, const float* __restrict__ regPart,
             float* __restrict__ out, int B) {
    __shared__ float lds[THREADS / 32];
    float L = 0.0f, R = 0.0f;
    for (int i = threadIdx.x; i < MAIN_BLOCKS; i += blockDim.x) L += lossPart[i];
    for (int i = threadIdx.x; i < REG_BLOCKS;  i += blockDim.x) R += regPart[i];
    float Ls = block_reduce(L, lds);
    float Rs = block_reduce(R, lds);
    if (threadIdx.x == 0) out[0] = -Ls / (float)B + Rs;
}

extern "C" void kernel_launch(void* const* d_in, const int* in_sizes, int n_in,
                              void* d_out, int out_size, void* d_ws, size_t ws_size,
                              hipStream_t stream) {
    const float* logits  = (const float*)d_in[0];
    const int*   targets = (const int*)d_in[1];     // harness int convention
    const float* weights = (const float*)d_in[2];
    const float* deltas  = (const float*)d_in[3];

    const int B   = in_sizes[1];          // 524288
    const int nw4 = in_sizes[2] / 4;      // weights as float4 count

    float* ws       = (float*)d_ws;
    float* lossPart = ws;                  // [0, MAIN_BLOCKS)
    float* regPart  = ws + MAIN_BLOCKS;    // [MAIN_BLOCKS, MAIN_BLOCKS+REG_BLOCKS)

    loss_kernel<<<MAIN_BLOCKS, THREADS, 0, stream>>>(logits, targets, lossPart, B);
    reg_kernel<<<REG_BLOCKS, THREADS, 0, stream>>>(weights, nw4, deltas, regPart);
    final_kernel<<<1, THREADS, 0, stream>>>(lossPart, regPart, (float*)d_out, B);
}